// Reconstruction_1039382085783
// MI455X (gfx1250) — compile-verified
//
#include <hip/hip_runtime.h>

// Inverse 2x2 Haar reconstruction: (N,4,512,512) f32 -> (N,1,1024,1024) f32.
// Memory-bound (268 MB traffic, ~0.13 GFLOP): optimize pure data movement.
// CDNA5 path: async HBM->LDS b128 copies (ASYNCcnt), conflict-free DS reads,
// b128 non-temporal stores.

typedef float v2f __attribute__((ext_vector_type(2)));
typedef float v4f __attribute__((ext_vector_type(4)));

#define S    512
#define SS   (S * S)         // 262144 floats per channel plane
#define OW   (2 * S)         // 1024

__global__ __launch_bounds__(256)
void haar_inv_kernel(const float* __restrict__ x, float* __restrict__ out) {
    __shared__ float lds[4 * S];          // 8 KB: 4 channels x 512 floats

    const int tid = threadIdx.x;          // 0..255
    const int b   = blockIdx.x;           // 0 .. N*S-1
    const int n   = b >> 9;               // b / 512
    const int h   = b & (S - 1);          // b % 512

    // Base of row h for channel 0 of image n (channel c is +c*SS).
    const float* xrow = x + (((size_t)n * 4) << 18) + ((size_t)h << 9);

    // ---- Stage: async copy the 4 channel row-segments into LDS (b128/lane).
    // 512 x 16B transactions per block; fully coalesced per channel.
#pragma unroll
    for (int i = 0; i < 2; ++i) {
        const int j  = (i << 8) + tid;    // 0..511
        const int c  = j >> 7;            // channel 0..3
        const int w4 = (j & 127) << 2;    // float offset 0,4,...,508
        const float* g = xrow + ((size_t)c << 18) + w4;
        float*       l = &lds[(c << 9) + w4];
#if defined(__gfx1250__)
        // LDS byte offset = low 32 bits of the flat shared-memory address
        unsigned           ldsOff = (unsigned)(unsigned long long)(const void*)l;
        unsigned long long gaddr  = (unsigned long long)(const void*)g;
        asm volatile("global_load_async_to_lds_b128 %0, %1, off"
                     :
                     : "v"(ldsOff), "v"(gaddr)
                     : "memory");
#else
        *(v4f*)l = *(const v4f*)g;        // fallback path (non-gfx1250 pass)
#endif
    }
#if defined(__gfx1250__)
    asm volatile("s_wait_asynccnt 0" ::: "memory");
#endif
    __syncthreads();

    // ---- Compute: each thread butterflies input columns w0, w0+1.
    // DS b64 reads: byte addr = c*2048 + 8*tid -> wave32 covers all 64 banks
    // exactly once per channel (conflict-free).
    const int w0 = tid << 1;
    const v2f x0 = *(const v2f*)&lds[          w0];
    const v2f x1 = *(const v2f*)&lds[    S   + w0];
    const v2f x2 = *(const v2f*)&lds[2 * S   + w0];
    const v2f x3 = *(const v2f*)&lds[3 * S   + w0];

    const v2f a  = x0 + x1;               // row-even pre-sum
    const v2f bb = x0 - x1;               // row-odd  pre-sum
    const v2f cc = x2 + x3;
    const v2f dd = x2 - x3;

    v4f r0, r1;                           // output rows 2h and 2h+1, cols 4*tid..4*tid+3
    r0.x = 0.5f * (a.x  + cc.x);  r0.y = 0.5f * (a.x  - cc.x);
    r0.z = 0.5f * (a.y  + cc.y);  r0.w = 0.5f * (a.y  - cc.y);
    r1.x = 0.5f * (bb.x + dd.x);  r1.y = 0.5f * (bb.x - dd.x);
    r1.z = 0.5f * (bb.y + dd.y);  r1.w = 0.5f * (bb.y - dd.y);

    // ---- Store: two 16B non-temporal stores, coalesced across the block.
    float* o0 = out + ((size_t)n << 20) + ((size_t)(2 * h) << 10) + (tid << 2);
    __builtin_nontemporal_store(r0, (v4f*)o0);
    __builtin_nontemporal_store(r1, (v4f*)(o0 + OW));
}

extern "C" void kernel_launch(void* const* d_in, const int* in_sizes, int n_in,
                              void* d_out, int out_size, void* d_ws, size_t ws_size,
                              hipStream_t stream) {
    (void)n_in; (void)out_size; (void)d_ws; (void)ws_size;
    const float* x   = (const float*)d_in[0];   // (N,4,512,512) f32
    float*       out = (float*)d_out;           // (N,1,1024,1024) f32
    const int N = in_sizes[0] / (4 * SS);       // 32
    haar_inv_kernel<<<dim3(N * S), dim3(256), 0, stream>>>(x, out);
}